// HomoGNN_35184372088984
// MI455X (gfx1250) — compile-verified
//
#include <hip/hip_runtime.h>
#include <math.h>

typedef __attribute__((ext_vector_type(2))) float v2f;
typedef __attribute__((ext_vector_type(8))) float v8f;

#define NNODES 100000
#define NEDGES 1600000
#define FIN 128
#define FH 16
#define FOUT 300
#define BN_EPS 1e-5f
#define NEG_INF (-3.402823466e38f)

// ---------------------------------------------------------------- utilities
__global__ void zero_kernel(float* __restrict__ p, int n) {
    int i = blockIdx.x * blockDim.x + threadIdx.x;
    if (i < n) p[i] = 0.0f;
}

__global__ void deg_count_kernel(const int* __restrict__ dst,
                                 float* __restrict__ deg, int E) {
    int e = blockIdx.x * blockDim.x + threadIdx.x;
    if (e < E) atomicAdd(&deg[dst[e]], 1.0f);
}

__global__ void deg_finish_kernel(const float* __restrict__ deg,
                                  float* __restrict__ dinv,
                                  float* __restrict__ invdeg, int n) {
    int i = blockIdx.x * blockDim.x + threadIdx.x;
    if (i < n) {
        float d = deg[i] + 1.0f;          // self loop
        dinv[i] = rsqrtf(d);
        invdeg[i] = 1.0f / d;
    }
}

// ------------------------------------------------- GEMM1: h0 = x @ W1 (WMMA)
// One wave per 16-row tile. K = 128 -> 32 steps of v_wmma_f32_16x16x4_f32.
__global__ __launch_bounds__(256) void gemm1_wmma_kernel(
    const float* __restrict__ x, const float* __restrict__ W1,
    float* __restrict__ h0, int ntiles) {
    int wave = threadIdx.x >> 5;
    int lane = threadIdx.x & 31;
    int tile = blockIdx.x * (blockDim.x >> 5) + wave;
    if (tile >= ntiles) return;            // wave-uniform: EXEC stays all-1s

    int m0    = tile * 16;
    int arow  = m0 + (lane & 15);          // A: M striped over lanes 0..15
    int khalf = (lane >> 4) * 2;           // lanes 16..31 hold K+2,K+3
    int col   = lane & 15;                 // B/D: N striped over lanes

    v8f c = {};
    for (int k0 = 0; k0 < FIN; k0 += 4) {
        int klo = k0 + khalf;
        v2f a, b;
        a.x = x[arow * FIN + klo];
        a.y = x[arow * FIN + klo + 1];
        b.x = W1[klo * FH + col];
        b.y = W1[(klo + 1) * FH + col];
        c = __builtin_amdgcn_wmma_f32_16x16x4_f32(
                false, a, false, b, (short)0, c, false, false);
    }
    // D layout: vgpr v -> row m0 + v + 8*(lane>=16), col = lane%16
    int rbase = m0 + ((lane >> 4) << 3);
#pragma unroll
    for (int v = 0; v < 8; ++v)
        h0[(rbase + v) * FH + col] = c[v];
}

// ------------------------------------------- edge scatter: agg[dst] += h[src]*coef
__global__ void scatter_kernel(const int* __restrict__ src,
                               const int* __restrict__ dst,
                               const float* __restrict__ dinv,
                               const float* __restrict__ hin,
                               float* __restrict__ agg, int E) {
    int e = blockIdx.x * blockDim.x + threadIdx.x;
    if (e >= E) return;
    int s = src[e], d = dst[e];
    float coef = dinv[s] * dinv[d];
    const float4* hp = (const float4*)(hin + (size_t)s * FH);
    float* op = agg + (size_t)d * FH;
#pragma unroll
    for (int i = 0; i < 4; ++i) {
        float4 v = hp[i];
        atomicAdd(op + 4 * i + 0, v.x * coef);
        atomicAdd(op + 4 * i + 1, v.y * coef);
        atomicAdd(op + 4 * i + 2, v.z * coef);
        atomicAdd(op + 4 * i + 3, v.w * coef);
    }
}

// --------------------------- layer1 tail: self-loop + bias + BatchNorm(eval)
__global__ void finish1_bn_kernel(const float* __restrict__ agg1,
                                  const float* __restrict__ h0,
                                  const float* __restrict__ invdeg,
                                  const float* __restrict__ b1,
                                  const float* __restrict__ gamma,
                                  const float* __restrict__ beta,
                                  const float* __restrict__ mean,
                                  const float* __restrict__ var,
                                  float* __restrict__ h_ws,
                                  float* __restrict__ h_out, int n16) {
    int i = blockIdx.x * blockDim.x + threadIdx.x;
    if (i >= n16) return;
    int node = i >> 4;
    int f = i & 15;
    float v = agg1[i] + h0[i] * invdeg[node] + b1[f];
    v = (v - mean[f]) * (rsqrtf(var[f] + BN_EPS) * gamma[f]) + beta[f];
    h_ws[i] = v;
    h_out[i] = v;
}

// ---------------- GEMM2 (+bias) with fused log_softmax, WMMA f32 16x16x4
// One block (8 waves) per 16-row tile. A = agg2 + h*invdeg (self-loop folded
// into the A-operand load). 19 N-tiles x 4 K-steps -> LDS -> row log-softmax.
__global__ __launch_bounds__(256) void gemm2_softmax_kernel(
    const float* __restrict__ agg2, const float* __restrict__ h,
    const float* __restrict__ invdeg, const float* __restrict__ W2,
    const float* __restrict__ b2, float* __restrict__ out) {
    __shared__ float slog[16][FOUT + 4];   // 16 x 304
    __shared__ float sred[16][16];

    int wave = threadIdx.x >> 5;
    int lane = threadIdx.x & 31;
    int m0 = blockIdx.x * 16;

    int arow  = m0 + (lane & 15);
    int khalf = (lane >> 4) * 2;
    int colq  = lane & 15;

    // Preload A (shared across all 19 N-tiles): 4 k-steps of 2 values/lane.
    float invd = invdeg[arow];
    v2f a[4];
#pragma unroll
    for (int kk = 0; kk < 4; ++kk) {
        int klo = kk * 4 + khalf;
        a[kk].x = agg2[arow * FH + klo] + h[arow * FH + klo] * invd;
        a[kk].y = agg2[arow * FH + klo + 1] + h[arow * FH + klo + 1] * invd;
    }

    for (int t = wave; t < 19; t += 8) {   // wave-uniform loop bound
        int col = t * 16 + colq;
        v8f c = {};
#pragma unroll
        for (int kk = 0; kk < 4; ++kk) {
            int klo = kk * 4 + khalf;
            v2f b;
            b.x = (col < FOUT) ? W2[klo * FOUT + col] : 0.0f;       // select, no branch
            b.y = (col < FOUT) ? W2[(klo + 1) * FOUT + col] : 0.0f;
            c = __builtin_amdgcn_wmma_f32_16x16x4_f32(
                    false, a[kk], false, b, (short)0, c, false, false);
        }
        if (col < FOUT) {
            float bias = b2[col];
            int rb = (lane >> 4) * 8;
#pragma unroll
            for (int v = 0; v < 8; ++v)
                slog[rb + v][col] = c[v] + bias;
        }
    }
    __syncthreads();

    // log_softmax over 300 cols: 16 threads cooperate per row.
    int row = threadIdx.x >> 4;
    int j = threadIdx.x & 15;

    float m = NEG_INF;
    for (int cI = j; cI < FOUT; cI += 16) m = fmaxf(m, slog[row][cI]);
    sred[row][j] = m;
    __syncthreads();
    float rm = sred[row][0];
#pragma unroll
    for (int k = 1; k < 16; ++k) rm = fmaxf(rm, sred[row][k]);
    __syncthreads();

    float ssum = 0.0f;
    for (int cI = j; cI < FOUT; cI += 16) ssum += __expf(slog[row][cI] - rm);
    sred[row][j] = ssum;
    __syncthreads();
    float tot = 0.0f;
#pragma unroll
    for (int k = 0; k < 16; ++k) tot += sred[row][k];
    float lse = rm + __logf(tot);

    size_t gr = (size_t)(m0 + row);
    for (int cI = j; cI < FOUT; cI += 16)
        out[gr * FOUT + cI] = slog[row][cI] - lse;
}

// ---------------------------------------------------------------- launcher
extern "C" void kernel_launch(void* const* d_in, const int* in_sizes, int n_in,
                              void* d_out, int out_size, void* d_ws, size_t ws_size,
                              hipStream_t stream) {
    (void)in_sizes; (void)n_in; (void)out_size; (void)ws_size;

    const float* x        = (const float*)d_in[0];
    const int*   eidx     = (const int*)d_in[1];
    const float* W1       = (const float*)d_in[2];
    const float* b1       = (const float*)d_in[3];
    const float* gamma    = (const float*)d_in[4];
    const float* beta     = (const float*)d_in[5];
    const float* run_mean = (const float*)d_in[6];
    const float* run_var  = (const float*)d_in[7];
    const float* W2       = (const float*)d_in[8];
    const float* b2       = (const float*)d_in[9];

    const int* src = eidx;            // edge_index[0]
    const int* dst = eidx + NEDGES;   // edge_index[1]

    float* ws     = (float*)d_ws;
    float* deg    = ws;                        // N
    float* dinv   = deg + NNODES;              // N
    float* invdeg = dinv + NNODES;             // N
    float* h0     = invdeg + NNODES;           // 16N
    float* agg1   = h0 + 16 * NNODES;          // 16N
    float* hbuf   = agg1 + 16 * NNODES;        // 16N
    float* agg2   = hbuf + 16 * NNODES;        // 16N

    float* out_h  = (float*)d_out;             // [N,16]
    float* out_ls = out_h + 16 * NNODES;       // [N,300]

    const int T = 256;
    const int ntiles = NNODES / 16;            // 6250 (exact)

    zero_kernel<<<(NNODES + T - 1) / T, T, 0, stream>>>(deg, NNODES);
    zero_kernel<<<(16 * NNODES + T - 1) / T, T, 0, stream>>>(agg1, 16 * NNODES);
    zero_kernel<<<(16 * NNODES + T - 1) / T, T, 0, stream>>>(agg2, 16 * NNODES);

    deg_count_kernel<<<(NEDGES + T - 1) / T, T, 0, stream>>>(dst, deg, NEDGES);
    deg_finish_kernel<<<(NNODES + T - 1) / T, T, 0, stream>>>(deg, dinv, invdeg, NNODES);

    gemm1_wmma_kernel<<<(ntiles + 7) / 8, T, 0, stream>>>(x, W1, h0, ntiles);

    scatter_kernel<<<(NEDGES + T - 1) / T, T, 0, stream>>>(src, dst, dinv, h0, agg1, NEDGES);

    finish1_bn_kernel<<<(16 * NNODES + T - 1) / T, T, 0, stream>>>(
        agg1, h0, invdeg, b1, gamma, beta, run_mean, run_var,
        hbuf, out_h, 16 * NNODES);

    scatter_kernel<<<(NEDGES + T - 1) / T, T, 0, stream>>>(src, dst, dinv, hbuf, agg2, NEDGES);

    gemm2_softmax_kernel<<<ntiles, T, 0, stream>>>(agg2, hbuf, invdeg, W2, b2, out_ls);
}